// MultiScaleWalkSampler_12275016532655
// MI455X (gfx1250) — compile-verified
//
#include <hip/hip_runtime.h>
#include <hip/hip_bf16.h>
#include <stdint.h>

// Temporal-biased random-walk sampler for MI455X (gfx1250, wave32).
// One wave32 == one walk. D = 64 neighbors -> 2 per lane.
// Row gather (64 f32 times + 64 i32 ids = 512B) staged into LDS with a single
// per-wave GLOBAL_LOAD_ASYNC_TO_LDS_B128 (lanes 0..15: times row, lanes 16..31:
// ids row, 16B per lane), completed with s_wait_asynccnt 0. Reductions are
// wave32 shfl_xor butterflies. Tables (51MB) are L2-resident (192MB L2), so
// this is an L2-latency-bound gather workload; WMMA is inapplicable (no GEMM).

#define WAVE 32
#define WAVES_PER_BLOCK 8
#define BLOCK (WAVE * WAVES_PER_BLOCK)
#define DDEG 64
#define INV_TEMP 10.0f   // 1 / TEMPERATURE, TEMPERATURE = 0.1
#define EPSW 1e-10f

#if __has_builtin(__builtin_amdgcn_global_load_async_to_lds_b128) && \
    __has_builtin(__builtin_amdgcn_s_wait_asynccnt)
#define HAVE_ASYNC_LDS 1
#else
#define HAVE_ASYNC_LDS 0
#endif

// Matches the builtin's parameter type exactly as reported by clang:
// "__attribute__((__vector_size__(4 * sizeof(int)))) int __device__ *"
typedef int v4i __attribute__((vector_size(16)));
typedef __attribute__((address_space(1))) v4i* gptr_v4i;  // global (AS1)
typedef __attribute__((address_space(3))) v4i* lptr_v4i;  // LDS (AS3)

__device__ __forceinline__ uint32_t pcg32(uint32_t x) {
  uint32_t s = x * 747796405u + 2891336453u;
  uint32_t w = ((s >> ((s >> 28u) + 4u)) ^ s) * 277803737u;
  return (w >> 22u) ^ w;
}

// Gumbel(0,1) sample from 24 random bits: g = -log(-log(u)), u in (0,1).
__device__ __forceinline__ float gumbel_from_bits(uint32_t h) {
  float u = (float)(h >> 8) * (1.0f / 16777216.0f) + 1e-7f;
  float nl = -__logf(u);
  return -__logf(fmaxf(nl, 1e-30f));
}

__global__ void __launch_bounds__(BLOCK)
temporal_walk_kernel(const int* __restrict__ src_nodes,
                     const float* __restrict__ src_times,
                     const int* __restrict__ nb_ids,
                     const float* __restrict__ nb_times,
                     const int* __restrict__ p_num_walks,
                     const int* __restrict__ p_walk_len,
                     float* __restrict__ out,
                     int B) {
  const int W = *p_num_walks;   // scalar inputs live in device memory
  const int L = *p_walk_len;
  const long BWL = (long)B * (long)W * (long)L;
  const int lane      = threadIdx.x & (WAVE - 1);
  const int waveInBlk = (int)(threadIdx.x >> 5);
  const int waveId    = (int)((blockIdx.x * blockDim.x + threadIdx.x) >> 5);
  const int nWaves    = (int)((gridDim.x * blockDim.x) >> 5);
  const int totalWalks = B * W;

  // Per-wave staging: [0..63] = times row (f32 bits), [64..127] = ids row.
  __shared__ uint32_t lds[WAVES_PER_BLOCK * 2 * DDEG];
  uint32_t* myLds = &lds[waveInBlk * 2 * DDEG];

  for (int walk = waveId; walk < totalWalks; walk += nWaves) {
    const int b = walk / W;
    int   curr_node = src_nodes[b];
    float curr_time = src_times[b];
    bool  alive = true;

    if (lane == 0) {
      const long o = (long)walk * L;
      out[o]           = (float)curr_node;   // step 0: the source itself
      out[BWL + o]     = curr_time;
      out[2 * BWL + o] = 1.0f;
    }

    for (int step = 1; step < L; ++step) {
      if (alive) {
        const long row = (long)curr_node * DDEG;

        // ---- Stage the neighbor row into LDS (512B per wave) ----
#if HAVE_ASYNC_LDS
        {
          const void* gsrc = (lane < 16)
              ? (const void*)(nb_times + row + (long)lane * 4)
              : (const void*)(nb_ids   + row + (long)(lane - 16) * 4);
          uint32_t* ldst = myLds + ((lane < 16) ? lane * 4 : DDEG + (lane - 16) * 4);
          __builtin_amdgcn_global_load_async_to_lds_b128(
              (gptr_v4i)(uintptr_t)gsrc,
              (lptr_v4i)(void*)ldst,
              /*offset=*/0, /*cpol=*/0);
          asm volatile("" ::: "memory");
          __builtin_amdgcn_s_wait_asynccnt(0);
          asm volatile("" ::: "memory");
        }
#else
        {
          // Fallback: coalesced VGPR loads, then write-through LDS so the
          // post-argmax winner lookup stays uniform.
          myLds[lane]             = __float_as_uint(nb_times[row + lane]);
          myLds[lane + 32]        = __float_as_uint(nb_times[row + lane + 32]);
          myLds[DDEG + lane]      = (uint32_t)nb_ids[row + lane];
          myLds[DDEG + lane + 32] = (uint32_t)nb_ids[row + lane + 32];
          asm volatile("" ::: "memory");
        }
#endif
        // ---- Per-lane pair of neighbors: j0 = lane, j1 = lane + 32 ----
        const float t0 = __uint_as_float(myLds[lane]);
        const float t1 = __uint_as_float(myLds[lane + 32]);
        const bool  m0 = t0 < curr_time;
        const bool  m1 = t1 < curr_time;

        // t_max over all 64 times (reference maxes over ALL, not only valid)
        float tm = fmaxf(t0, t1);
#pragma unroll
        for (int off = 16; off > 0; off >>= 1)
          tm = fmaxf(tm, __shfl_xor(tm, off, WAVE));
        tm = (tm > 0.0f) ? tm : 1.0f;

        // w = exp((t - tmax)/tau) * mask ; sum over 64
        const float w0 = m0 ? __expf((t0 - tm) * INV_TEMP) : 0.0f;
        const float w1 = m1 ? __expf((t1 - tm) * INV_TEMP) : 0.0f;
        float s = w0 + w1;
#pragma unroll
        for (int off = 16; off > 0; off >>= 1)
          s += __shfl_xor(s, off, WAVE);
        const float inv = 1.0f / (s + EPSW);

        // Gumbel-argmax over log(probs + eps) + g
        const uint32_t base =
            pcg32((uint32_t)walk * 0x9E3779B9u ^ ((uint32_t)step * 0x85EBCA77u));
        const float sc0 = __logf(w0 * inv + EPSW) +
                          gumbel_from_bits(pcg32(base + (uint32_t)lane));
        const float sc1 = __logf(w1 * inv + EPSW) +
                          gumbel_from_bits(pcg32(base + (uint32_t)lane + 32u));

        float bs; int bj;
        if (sc0 >= sc1) { bs = sc0; bj = lane; }
        else            { bs = sc1; bj = lane + 32; }
#pragma unroll
        for (int off = 16; off > 0; off >>= 1) {
          const float os = __shfl_xor(bs, off, WAVE);
          const int   oj = __shfl_xor(bj, off, WAVE);
          // strictly-greater wins; ties go to the lower index (argmax semantics)
          if (os > bs || (os == bs && oj < bj)) { bs = os; bj = oj; }
        }

        if (__any(m0 || m1)) {
          curr_time = __uint_as_float(myLds[bj]);        // winner's time
          curr_node = (int)myLds[DDEG + bj];             // winner's node id
        } else {
          alive = false;                                  // sticky termination
        }
      }

      if (lane == 0) {
        const long o = (long)walk * L + step;
        out[o]           = (float)curr_node;   // ids < 2^24 -> exact in f32
        out[BWL + o]     = curr_time;
        out[2 * BWL + o] = alive ? 1.0f : 0.0f;
      }
    }
  }
}

extern "C" void kernel_launch(void* const* d_in, const int* in_sizes, int n_in,
                              void* d_out, int out_size, void* d_ws, size_t ws_size,
                              hipStream_t stream) {
  (void)n_in; (void)d_ws; (void)ws_size; (void)out_size;
  const int*   src_nodes = (const int*)d_in[0];
  const float* src_times = (const float*)d_in[1];
  const int*   nb_ids    = (const int*)d_in[2];
  const float* nb_times  = (const float*)d_in[3];
  const int*   p_W       = (const int*)d_in[4];   // num_walks (device scalar)
  const int*   p_L       = (const int*)d_in[5];   // walk_length (device scalar)
  float* out = (float*)d_out;
  const int B = in_sizes[0];

  // Grid-stride over B*W walks (W only known on device); 16K wave32s keeps
  // every SIMD32 saturated to hide L2 gather latency.
  const int blocks = 2048;
  temporal_walk_kernel<<<blocks, BLOCK, 0, stream>>>(
      src_nodes, src_times, nb_ids, nb_times, p_W, p_L, out, B);
}